// BatchedLIDIA_5995774345912
// MI455X (gfx1250) — compile-verified
//
#include <hip/hip_runtime.h>
#include <hip/hip_bf16.h>

typedef __attribute__((ext_vector_type(2))) float v2f;
typedef __attribute__((ext_vector_type(8))) float v8f;
typedef __attribute__((ext_vector_type(4))) unsigned int u32x4;
typedef __attribute__((ext_vector_type(8))) int i32x8;
typedef __attribute__((ext_vector_type(4))) int i32x4;

#define H 192
#define W 192
#define NPIX (H * W)
#define WS 29
#define NOFF (WS * WS)
#define KNN 14
#define PS 5

__device__ __forceinline__ int refl(int i, int n) {
    if (i < 0) i = -i;
    if (i >= n) i = 2 * n - 2 - i;
    return i;
}
__device__ __forceinline__ int imin(int a, int b) { return a < b ? a : b; }

// ---------------------------------------------------------------------------
// Stage 1: per-channel means of normalized image
// ---------------------------------------------------------------------------
__global__ void mean_kernel(const float* __restrict__ noisy, float* __restrict__ means) {
    __shared__ float red[256];
    const int c = blockIdx.x;
    float s = 0.f;
    for (int i = threadIdx.x; i < NPIX; i += 256) s += noisy[c * NPIX + i];
    red[threadIdx.x] = s;
    __syncthreads();
    for (int st = 128; st > 0; st >>= 1) {
        if (threadIdx.x < st) red[threadIdx.x] += red[threadIdx.x + st];
        __syncthreads();
    }
    if (threadIdx.x == 0)
        means[c] = (red[0] / ((float)NPIX * 255.f) - 0.5f) * 2.0f;
}

// x = ((noisy/255 - .5)*2) - mean_c ; gray = sum_c x*gw
__global__ void normalize_kernel(const float* __restrict__ noisy, const float* __restrict__ gw,
                                 const float* __restrict__ means, float* __restrict__ x,
                                 float* __restrict__ gray) {
    const int i = blockIdx.x * blockDim.x + threadIdx.x;
    if (i >= NPIX) return;
    float g = 0.f;
#pragma unroll
    for (int c = 0; c < 3; ++c) {
        float xv = (noisy[c * NPIX + i] / 255.f - 0.5f) * 2.0f - means[c];
        x[c * NPIX + i] = xv;
        g += xv * gw[c];
    }
    gray[i] = g;
}

// gray_s = 3x3 bilinear on reflect-padded gray
__global__ void smooth_kernel(const float* __restrict__ gray, const float* __restrict__ bw,
                              float* __restrict__ grays) {
    const int i = blockIdx.x * blockDim.x + threadIdx.x;
    if (i >= NPIX) return;
    const int y = i / W, x = i % W;
    float s = 0.f;
#pragma unroll
    for (int di = 0; di < 3; ++di)
#pragma unroll
        for (int dj = 0; dj < 3; ++dj)
            s += bw[di * 3 + dj] * gray[refl(y - 1 + di, H) * W + refl(x - 1 + dj, W)];
    grays[i] = s;
}

__global__ void zero_kernel(float* __restrict__ p, int n) {
    const int i = blockIdx.x * blockDim.x + threadIdx.x;
    if (i < n) p[i] = 0.f;
}

// ---------------------------------------------------------------------------
// Stage 2 (per scale): reflect pad gray + x into Gp [Hp*Hp], Np [3*Hp*Hp]
// ---------------------------------------------------------------------------
__global__ void pad_kernel(const float* __restrict__ gsrc, const float* __restrict__ x,
                           float* __restrict__ Gp, float* __restrict__ Np, int Hp, int P) {
    const int i = blockIdx.x * blockDim.x + threadIdx.x;
    if (i >= Hp * Hp) return;
    const int u = i / Hp, v = i % Hp;
    const int gy = refl(u - P, H), gx = refl(v - P, W);
    Gp[i] = gsrc[gy * W + gx];
#pragma unroll
    for (int c = 0; c < 3; ++c)
        Np[c * Hp * Hp + i] = x[c * NPIX + gy * W + gx];
}

// ---------------------------------------------------------------------------
// Horizontal 5-tap dilated box filter of Gp^2 via V_WMMA_F32_16X16X4_F32.
// t1[u][v] = sum_{j=0..4} Gp[u][v+j*DIL]^2   (v < Wt = Hp - 4*DIL = 220)
// D = A x B: A[m,k] = g^2[y0+m][x0+c0+k], B[k,n] = banded 0/1 selection.
// One wave (32 threads) per 16x16 output tile; K covered in chunks of 4.
// ---------------------------------------------------------------------------
template <int DIL>
__global__ void hbox_wmma_kernel(const float* __restrict__ Gp, float* __restrict__ t1,
                                 int Hp, int Wt) {
    const int x0 = blockIdx.x * 16;
    const int y0 = blockIdx.y * 16;
    const int lane = threadIdx.x;   // 0..31
    const int m = lane & 15;        // A row
    const int kb = (lane >> 4) * 2; // local K pair base
    const int n = lane & 15;        // B column
    constexpr int SPAN = 16 + 4 * DIL;
    constexpr int NCH = (SPAN + 3) / 4;

#if __has_builtin(__builtin_amdgcn_wmma_f32_16x16x4_f32)
    const int yr = imin(y0 + m, Hp - 1);
    v8f acc = {};
#pragma unroll
    for (int ch = 0; ch < NCH; ++ch) {
        const int c0 = ch * 4;
        const int ca = imin(x0 + c0 + kb, Hp - 1);
        const int cb = imin(x0 + c0 + kb + 1, Hp - 1);
        float ga = Gp[yr * Hp + ca];
        float gb = Gp[yr * Hp + cb];
        v2f a;
        a.x = ga * ga;
        a.y = gb * gb;
        // banded selection matrix S[c][n] = 1 iff c-n in {0, DIL, .., 4*DIL}
        int d0 = c0 + kb - n;
        int d1 = c0 + kb + 1 - n;
        v2f b;
        b.x = (d0 >= 0 && d0 <= 4 * DIL && (d0 % DIL) == 0) ? 1.0f : 0.0f;
        b.y = (d1 >= 0 && d1 <= 4 * DIL && (d1 % DIL) == 0) ? 1.0f : 0.0f;
        acc = __builtin_amdgcn_wmma_f32_16x16x4_f32(false, a, false, b, (short)0, acc,
                                                    false, false);
    }
    // D layout: VGPR v -> M = v + (lane>=16 ? 8 : 0), N = lane&15
#pragma unroll
    for (int v = 0; v < 8; ++v) {
        const int row = y0 + v + ((lane >> 4) * 8);
        const int col = x0 + n;
        if (row < Hp && col < Wt) t1[row * Wt + col] = acc[v];
    }
#else
    // scalar fallback: 32 threads cover the 16x16 tile in 8 row-steps
    const int half = lane >> 4;
#pragma unroll
    for (int v = 0; v < 8; ++v) {
        const int row = y0 + v + half * 8;
        const int col = x0 + n;
        if (row < Hp && col < Wt) {
            float s = 0.f;
#pragma unroll
            for (int j = 0; j < PS; ++j) {
                float g = Gp[row * Hp + col + j * DIL];
                s += g * g;
            }
            t1[row * Wt + col] = s;
        }
    }
#endif
}

// vertical pass: SS[u][v] = sum_{i=0..4} t1[u+i*DIL][v], 220x220
template <int DIL>
__global__ void vbox_kernel(const float* __restrict__ t1, float* __restrict__ SS) {
    const int i = blockIdx.x * blockDim.x + threadIdx.x;
    if (i >= 220 * 220) return;
    const int u = i / 220, v = i % 220;
    float s = 0.f;
#pragma unroll
    for (int j = 0; j < PS; ++j) s += t1[(u + j * DIL) * 220 + v];
    SS[i] = s;
}

// ---------------------------------------------------------------------------
// TDM: DMA a 2D tile (tile_h x tile_w f32, row-major packed) from global into LDS.
// D# built per CDNA5 ISA; clang-23 6-arg builtin:
//   (uint32x4 g0, int32x8 g1, int32x4 g2, int32x4 g3, int32x8 extra, i32 cpol)
// ---------------------------------------------------------------------------
__device__ __forceinline__ void tdm_load_tile_2d(unsigned lds_off, const float* gptr,
                                                 int tensor_w, int tensor_h, int tile_w,
                                                 int tile_h) {
#if __has_builtin(__builtin_amdgcn_tensor_load_to_lds)
    const unsigned long long ga = (unsigned long long)(size_t)gptr;
    u32x4 g0;
    g0[0] = 1u;  // count=1, user descriptor, no gather
    g0[1] = lds_off;
    g0[2] = (unsigned)(ga & 0xFFFFFFFFull);
    g0[3] = (unsigned)((ga >> 32) & 0x1FFFFFFull) | (2u << 30);  // type=2 (image)
    const unsigned long long s0 = (unsigned long long)tensor_w;
    const unsigned long long s1 = (unsigned long long)tensor_w * (unsigned long long)tensor_h;
    i32x8 g1;
    g1[0] = (int)(2u << 16);  // data_size = 4B; no multicast/pad/iterate
    g1[1] = (int)(((unsigned)tensor_w & 0xFFFFu) << 16);                 // dim0[15:0]
    g1[2] = (int)((((unsigned)tensor_w >> 16) & 0xFFFFu) |
                  (((unsigned)tensor_h & 0xFFFFu) << 16));               // dim0[31:16], dim1[15:0]
    g1[3] = (int)((((unsigned)tensor_h >> 16) & 0xFFFFu) |
                  (((unsigned)tile_w & 0xFFFFu) << 16));                 // dim1[31:16], tile0
    g1[4] = (int)((unsigned)tile_h & 0xFFFFu);                           // tile1, tile2=0
    g1[5] = (int)(unsigned)(s0 & 0xFFFFFFFFull);                         // stride0[31:0]
    g1[6] = (int)((unsigned)((s0 >> 32) & 0xFFFFull) |
                  ((unsigned)(s1 & 0xFFFFull) << 16));                   // stride0 hi, stride1 lo
    g1[7] = (int)(unsigned)((s1 >> 16) & 0xFFFFFFFFull);                 // stride1 rest
    i32x4 z4 = {};
    i32x8 z8 = {};
    __builtin_amdgcn_tensor_load_to_lds(g0, g1, z4, z4, z8, 0);
#else
    (void)lds_off; (void)gptr; (void)tensor_w; (void)tensor_h; (void)tile_w; (void)tile_h;
#endif
}

// ---------------------------------------------------------------------------
// dist + topK + softmax, one thread per pixel, 16x16 pixel tile per block.
// dist(o) = SSc + SS[y+oy][x+ox] - 2 * <center_patch, patch(y+oy,x+ox)>
// Search-window tile of Gp staged in LDS via Tensor Data Mover.
// ---------------------------------------------------------------------------
template <int DIL>
__global__ void dist_topk_kernel(const float* __restrict__ Gp, const float* __restrict__ SS,
                                 float* __restrict__ topW, int* __restrict__ topIdx,
                                 const int* __restrict__ sigma_p) {
    constexpr int HP = (DIL == 1) ? 224 : 228;
    constexpr int T = 44 + 4 * DIL;  // 48 (dil=1) / 52 (dil=2)
    __shared__ float tile[T * T];

    const int x0 = blockIdx.x * 16;
    const int y0 = blockIdx.y * 16;
    const int tid = threadIdx.x;  // 0..255

#if __has_builtin(__builtin_amdgcn_tensor_load_to_lds)
    if (tid == 0) {
        tdm_load_tile_2d((unsigned)(size_t)tile, Gp + y0 * HP + x0, HP, HP, T, T);
        __builtin_amdgcn_s_wait_tensorcnt(0);
    }
#else
    for (int j = tid; j < T * T; j += 256) {
        const int r = j / T, cc = j % T;
        tile[j] = Gp[(y0 + r) * HP + (x0 + cc)];
    }
#endif
    __syncthreads();

    const int lx = tid & 15, ly = tid >> 4;
    const int x = x0 + lx, y = y0 + ly;

    // cache center patch (Gp at +WS/2 offset)
    float cp[PS * PS];
#pragma unroll
    for (int pi = 0; pi < PS; ++pi)
#pragma unroll
        for (int pj = 0; pj < PS; ++pj)
            cp[pi * PS + pj] = tile[(ly + 14 + DIL * pi) * T + (lx + 14 + DIL * pj)];

    const float SSc = SS[(y + 14) * 220 + (x + 14)];

    float d[KNN];
    int id[KNN];
#pragma unroll
    for (int k = 0; k < KNN; ++k) {
        d[k] = 3.0e38f;
        id[k] = 0;
    }

    for (int oy = 0; oy < WS; ++oy) {
        for (int ox = 0; ox < WS; ++ox) {
            float Xc = 0.f;
#pragma unroll
            for (int pi = 0; pi < PS; ++pi)
#pragma unroll
                for (int pj = 0; pj < PS; ++pj)
                    Xc = __builtin_fmaf(cp[pi * PS + pj],
                                        tile[(ly + oy + DIL * pi) * T + (lx + ox + DIL * pj)],
                                        Xc);
            const float dist = SSc + SS[(y + oy) * 220 + (x + ox)] - 2.f * Xc;
            if (dist < d[KNN - 1]) {
                d[KNN - 1] = dist;
                id[KNN - 1] = oy * WS + ox;
#pragma unroll
                for (int j = KNN - 1; j > 0; --j) {
                    if (d[j] < d[j - 1]) {
                        float td = d[j]; d[j] = d[j - 1]; d[j - 1] = td;
                        int ti = id[j]; id[j] = id[j - 1]; id[j - 1] = ti;
                    }
                }
            }
        }
    }

    const float sig01 = (float)(*sigma_p) / 255.0f / 0.5f;
    const float lam = (2.f * sig01) * (2.f * sig01) * (float)(PS * PS) + 1e-6f;
    float wv[KNN];
    float wsum = 0.f;
#pragma unroll
    for (int k = 0; k < KNN; ++k) {
        wv[k] = __expf((d[0] - d[k]) / lam);
        wsum += wv[k];
    }
    const int pix = y * W + x;
    const float inv = 1.0f / wsum;
#pragma unroll
    for (int k = 0; k < KNN; ++k) {
        topW[k * NPIX + pix] = wv[k] * inv;
        topIdx[k * NPIX + pix] = id[k];
    }
}

// ---------------------------------------------------------------------------
// Gather K weighted patches + overlap-add fold (expressed as gather-of-gather),
// accumulate this scale's denoised estimate into accum.
// ---------------------------------------------------------------------------
template <int DIL>
__global__ void gather_fold_kernel(const float* __restrict__ Np, const float* __restrict__ topW,
                                   const int* __restrict__ topIdx, float* __restrict__ accum) {
    constexpr int HP = (DIL == 1) ? 224 : 228;
    const int i = blockIdx.x * blockDim.x + threadIdx.x;
    if (i >= NPIX) return;
    const int y = i / W, x = i % W;
    float a0 = 0.f, a1 = 0.f, a2 = 0.f;
    int cnt = 0;
    for (int qi = -2; qi <= 2; ++qi) {
        const int ys = y + DIL * qi;
        if (ys < 0 || ys >= H) continue;
        for (int qj = -2; qj <= 2; ++qj) {
            const int xs = x + DIL * qj;
            if (xs < 0 || xs >= W) continue;
            ++cnt;
            const int pi = 2 - qi, pj = 2 - qj;
            const int ps = ys * W + xs;
#pragma unroll
            for (int k = 0; k < KNN; ++k) {
                const float wk = topW[k * NPIX + ps];
                const int idx = topIdx[k * NPIX + ps];
                const int di = idx / WS, dj = idx % WS;
                const int off = (ys + di + DIL * pi) * HP + (xs + dj + DIL * pj);
                a0 = __builtin_fmaf(wk, Np[off], a0);
                a1 = __builtin_fmaf(wk, Np[HP * HP + off], a1);
                a2 = __builtin_fmaf(wk, Np[2 * HP * HP + off], a2);
            }
        }
    }
    const float inv = 1.0f / (float)cnt;
    accum[i] += a0 * inv;
    accum[NPIX + i] += a1 * inv;
    accum[2 * NPIX + i] += a2 * inv;
}

// out = 255 * (0.25*(deno0+deno1) + 0.5*mean_c + 0.5)
__global__ void final_kernel(const float* __restrict__ accum, const float* __restrict__ means,
                             float* __restrict__ out) {
    const int i = blockIdx.x * blockDim.x + threadIdx.x;
    if (i >= NPIX) return;
#pragma unroll
    for (int c = 0; c < 3; ++c)
        out[c * NPIX + i] = 255.f * (0.25f * accum[c * NPIX + i] + 0.5f * means[c] + 0.5f);
}

// ---------------------------------------------------------------------------
extern "C" void kernel_launch(void* const* d_in, const int* in_sizes, int n_in,
                              void* d_out, int out_size, void* d_ws, size_t ws_size,
                              hipStream_t stream) {
    const float* noisy = (const float*)d_in[0];
    const float* gw = (const float*)d_in[1];
    const float* bw = (const float*)d_in[2];
    const int* sig = (const int*)d_in[3];
    float* out = (float*)d_out;

    // workspace carve-up (floats); total ~6.6 MB
    float* ws = (float*)d_ws;
    float* xbuf = ws;                       // 3*NPIX
    float* gray = xbuf + 3 * NPIX;          // NPIX
    float* grays = gray + NPIX;             // NPIX
    float* Gp = grays + NPIX;               // 228*228 (max)
    float* Np = Gp + 228 * 228;             // 3*228*228
    float* t1 = Np + 3 * 228 * 228;         // 228*220
    float* SS = t1 + 228 * 220;             // 220*220
    float* topW = SS + 220 * 220;           // KNN*NPIX
    int* topIdx = (int*)(topW + KNN * NPIX);  // KNN*NPIX ints
    float* accum = (float*)(topIdx + KNN * NPIX);  // 3*NPIX
    float* means = accum + 3 * NPIX;        // 3

    const int gpix = (NPIX + 255) / 256;

    mean_kernel<<<3, 256, 0, stream>>>(noisy, means);
    normalize_kernel<<<gpix, 256, 0, stream>>>(noisy, gw, means, xbuf, gray);
    smooth_kernel<<<gpix, 256, 0, stream>>>(gray, bw, grays);
    zero_kernel<<<(3 * NPIX + 255) / 256, 256, 0, stream>>>(accum, 3 * NPIX);

    // ---- scale 0: dil=1, P=16, Hp=224, search on gray ----
    pad_kernel<<<(224 * 224 + 255) / 256, 256, 0, stream>>>(gray, xbuf, Gp, Np, 224, 16);
    hbox_wmma_kernel<1><<<dim3(14, 14), 32, 0, stream>>>(Gp, t1, 224, 220);
    vbox_kernel<1><<<(220 * 220 + 255) / 256, 256, 0, stream>>>(t1, SS);
    dist_topk_kernel<1><<<dim3(12, 12), 256, 0, stream>>>(Gp, SS, topW, topIdx, sig);
    gather_fold_kernel<1><<<gpix, 256, 0, stream>>>(Np, topW, topIdx, accum);

    // ---- scale 1: dil=2, P=18, Hp=228, search on smoothed gray ----
    pad_kernel<<<(228 * 228 + 255) / 256, 256, 0, stream>>>(grays, xbuf, Gp, Np, 228, 18);
    hbox_wmma_kernel<2><<<dim3(14, 15), 32, 0, stream>>>(Gp, t1, 228, 220);
    vbox_kernel<2><<<(220 * 220 + 255) / 256, 256, 0, stream>>>(t1, SS);
    dist_topk_kernel<2><<<dim3(12, 12), 256, 0, stream>>>(Gp, SS, topW, topIdx, sig);
    gather_fold_kernel<2><<<gpix, 256, 0, stream>>>(Np, topW, topIdx, accum);

    final_kernel<<<gpix, 256, 0, stream>>>(accum, means, out);

    (void)in_sizes; (void)n_in; (void)out_size; (void)ws_size;
}